// NormalizedEKVConv2d_27144193310926
// MI455X (gfx1250) — compile-verified
//
#include <hip/hip_runtime.h>
#include <math.h>

#ifndef __has_builtin
#define __has_builtin(x) 0
#endif

typedef unsigned int v4u __attribute__((ext_vector_type(4)));
typedef int          v4i __attribute__((ext_vector_type(4)));
typedef int          v8i __attribute__((ext_vector_type(8)));

// phi = 2*1.5*0.025 = 0.075 ; scale to log2 domain
#define SCALE_Z 19.235933878519512f   // (1/phi) * log2(e)
#define DELTA_Z 9.6179669392597561f   // (VD/phi) * log2(e)
// tia_gain*ALPHA*ln2^2 = (500000/24)*2e-6*0.693147^2
#define C_OUT   0.020018875579925058f

__device__ __forceinline__ float fast_exp2(float v) {
#if defined(__HIP_DEVICE_COMPILE__) && __has_builtin(__builtin_amdgcn_exp2f)
  return __builtin_amdgcn_exp2f(v);   // v_exp_f32
#else
  return exp2f(v);
#endif
}
__device__ __forceinline__ float fast_log2(float v) {
#if defined(__HIP_DEVICE_COMPILE__) && __has_builtin(__builtin_amdgcn_logf)
  return __builtin_amdgcn_logf(v);    // v_log_f32 (base-2)
#else
  return log2f(v);
#endif
}

#if defined(__HIP_DEVICE_COMPILE__) && __has_builtin(__builtin_amdgcn_tensor_load_to_lds)
#define EKV_HAS_TDM 1
#else
#define EKV_HAS_TDM 0
#endif

// out[n,o,py,px] = C_OUT * sum_r [ log2(1+2^z1)^2 - log2(1+2^(z1-DELTA))^2 ],
//   z1 = x*S - w*S,  S = (1/phi)*log2(e)
__global__ __launch_bounds__(256) void ekv_conv_kernel(
    const float* __restrict__ x,      // [4,64,32,32]
    const float* __restrict__ theta,  // [64,576]
    float* __restrict__ out)          // [4,64,32,32]
{
  __shared__ float sW[8 * 576];       // 18 KB: 8 output channels x 576 fan-in (prescaled)

  const int tid = threadIdx.x;
  const int n   = blockIdx.z;
  const int o0  = blockIdx.y * 8;
  const int ty  = ((blockIdx.x >> 1) << 4) + (tid >> 4);   // pixel row 0..31
  const int tx  = ((blockIdx.x & 1) << 4) + (tid & 15);    // pixel col 0..31

#if EKV_HAS_TDM
  // ---- Tensor Data Mover: DMA theta[o0:o0+8][0:576] (contiguous 4608 f32) into LDS ----
  if (tid < 32) {  // one wave issues the TDM op (EXEC ignored by TDM; other waves execz-skip)
    unsigned long long ga = (unsigned long long)(const void*)(theta + (size_t)o0 * 576);
    unsigned lds_off      = (unsigned)(unsigned long long)(const void*)&sW[0];

    // D# group 0: count=1, lds_addr, global_addr[56:0], type=2
    v4u g0 = { 1u,
               lds_off,
               (unsigned)ga,
               (unsigned)((ga >> 32) & 0x01FFFFFFull) | 0x80000000u };

    // D# group 1: data_size=4B; tensor_dim0=4608; tensor_dim1=1; tile_dim0=4608;
    //             tile_dim1=1; tile_dim2=0(unused); stride0=4608; stride1=4608
    v8i g1 = { (int)(2u << 16),
               (int)((4608u & 0xFFFFu) << 16),
               (int)((4608u >> 16) | (1u << 16)),
               (int)(4608u << 16),
               (int)1,
               (int)4608,
               (int)(4608u << 16),
               0 };
    // groups 2/3: benign values for unused dims
    v4i g2 = { 1, 1, 4608, 0 };
    v4i g3 = { 4608, (int)(1u << 16), 0, 0 };

#if __clang_major__ >= 23
    v8i g4 = { 0, 0, 0, 0, 0, 0, 0, 0 };
    __builtin_amdgcn_tensor_load_to_lds(g0, g1, g2, g3, g4, 0);
#else
    __builtin_amdgcn_tensor_load_to_lds(g0, g1, g2, g3, 0);
#endif
    __builtin_amdgcn_s_wait_tensorcnt(0);
  }
  __syncthreads();
  asm volatile("" ::: "memory");
  // Prescale weights in place: w' = w * S
  for (int t = tid; t < 4608; t += 256) sW[t] *= SCALE_Z;
  __syncthreads();
#else
  // Fallback: staged copy with scale
  for (int t = tid; t < 4608; t += 256) sW[t] = theta[(size_t)o0 * 576 + t] * SCALE_Z;
  __syncthreads();
#endif

  const float* xb = x + ((size_t)n << 16);  // n*64*1024

  float acc[8];
#pragma unroll
  for (int u = 0; u < 8; ++u) acc[u] = 0.0f;

  for (int c = 0; c < 64; ++c) {
    // Load the 3x3 zero-padded patch for this pixel/channel, prescaled to log2 units.
    float xv[9];
#pragma unroll
    for (int i = 0; i < 3; ++i) {
#pragma unroll
      for (int j = 0; j < 3; ++j) {
        int yy = ty + i - 1;
        int xx = tx + j - 1;
        bool ok = ((unsigned)yy < 32u) & ((unsigned)xx < 32u);
        xv[i * 3 + j] = ok ? xb[(c << 10) + (yy << 5) + xx] * SCALE_Z : 0.0f;
      }
    }
    // 8-channel register blocking; weight reads are wave-uniform LDS broadcasts.
#pragma unroll
    for (int u = 0; u < 8; ++u) {
      const float* wp = &sW[u * 576 + c * 9];
      float a = acc[u];
#pragma unroll
      for (int q = 0; q < 9; ++q) {
        float z1 = xv[q] - wp[q];
        float ea = fast_exp2(z1);
        float eb = fast_exp2(z1 - DELTA_Z);
        float sa = fast_log2(1.0f + ea);   // softplus / ln2
        float sb = fast_log2(1.0f + eb);
        a = __builtin_fmaf(sa - sb, sa + sb, a);  // sa^2 - sb^2 accumulated
      }
      acc[u] = a;
    }
  }

  const int pix = (ty << 5) + tx;
#pragma unroll
  for (int u = 0; u < 8; ++u) {
    out[(((n << 6) + o0 + u) << 10) + pix] = acc[u] * C_OUT;
  }
}

extern "C" void kernel_launch(void* const* d_in, const int* in_sizes, int n_in,
                              void* d_out, int out_size, void* d_ws, size_t ws_size,
                              hipStream_t stream) {
  const float* x     = (const float*)d_in[0];   // 4*64*32*32
  const float* theta = (const float*)d_in[1];   // 64*64*3*3
  float* out         = (float*)d_out;           // 4*64*32*32
  (void)in_sizes; (void)n_in; (void)out_size; (void)d_ws; (void)ws_size;

  dim3 grid(4, 8, 4);   // pixel-tiles x o-blocks x batch
  dim3 block(256);      // 8 wave32s
  ekv_conv_kernel<<<grid, block, 0, stream>>>(x, theta, out);
}